// Multi_LSTM_Unit_76759655514947
// MI455X (gfx1250) — compile-verified
//
#include <hip/hip_runtime.h>
#include <math.h>

// Problem constants (match reference)
#define BB 64
#define TT 512
#define DD 512
#define UU 1024
#define NG 4096   // 4*U

typedef __attribute__((ext_vector_type(2))) float v2f;
typedef __attribute__((ext_vector_type(8))) float v8f;

__device__ __forceinline__ float sigmoidf_(float x) {
    return 1.0f / (1.0f + __expf(-x));
}

// Low 32 bits of a generic pointer to __shared__ = wave-relative LDS byte offset
__device__ __forceinline__ unsigned lds_off_(const void* p) {
    return (unsigned)(uintptr_t)p;
}

// CDNA5 async global->LDS copy, 16 bytes per lane (tracked by ASYNCcnt)
__device__ __forceinline__ void async_ld16_(unsigned lds_off, const float* g) {
    unsigned long long ga = (unsigned long long)(uintptr_t)g;
    asm volatile("global_load_async_to_lds_b128 %0, %1, off"
                 :: "v"(lds_off), "v"(ga) : "memory");
}

// One LSTM timestep: z = [x_t | h_prev] @ [W ; R] + bias   (M=64, N=4096, K=Kx+U)
// Grid: 64 WGs (one per 16-unit block), 256 threads = 8 waves:
//   wave w -> gate (w&3), K-half (w>>2). Each wave: 4 M-tiles x 16 cols, half the K range.
// A chunks staged via double-buffered async global->LDS loads; partial z's for the two
// K-halves are summed in LDS during the fused gate phase.
__global__ __launch_bounds__(256)
void lstm_step_kernel(const float* __restrict__ xt, size_t xStride, int Kx,
                      const float* __restrict__ hprev, size_t hStride,
                      const float* __restrict__ W, const float* __restrict__ R,
                      const float* __restrict__ bias,
                      float* __restrict__ cstate,
                      float* __restrict__ hout, size_t houtStride,
                      int first) {
    // Union: A staging [half][buf][64][36] (36864 B) overlaps z exchange [2][64][65]
    // (33280 B). z is only written after the K loop's final barrier.
    __shared__ __attribute__((aligned(16))) float smem[9216];
    float (*Ach)[2][64][36] = reinterpret_cast<float (*)[2][64][36]>(smem);
    float (*zsh)[64][65]    = reinterpret_cast<float (*)[64][65]>(smem);

    const int tid  = threadIdx.x;
    const int wv   = tid >> 5;
    const int lane = tid & 31;
    const int g    = wv & 3;             // gate id (i,f,g,o)
    const int hf   = wv >> 2;            // K-half id
    const int lc   = lane & 15;
    const int kksel = (lane >> 4) << 1;  // 0 or 2 (WMMA k-pair select)
    const int u0   = blockIdx.x * 16;
    const int ncol = g * UU + u0 + lc;   // global GEMM column

    const int Ktot  = first ? Kx : (Kx + UU);
    const int Khalf = Ktot >> 1;         // always a multiple of 32
    const int nch   = Khalf >> 5;        // 32-wide chunks per half

    // Cooperative async-load geometry: 256 threads move both halves' chunks,
    // 4 x b128 per thread per iteration (2 rows for each half).
    const int rowA = tid >> 3;           // 0..31
    const int colL = (tid & 7) << 2;     // 0,4,...,28

    auto issue = [&](int c, int buf) {
        const int k0 = c << 5;
#pragma unroll
        for (int h2 = 0; h2 < 2; ++h2) {
            const int kg = h2 * Khalf + k0 + colL;   // segment uniform per (h2, chunk)
            const float* sb;  size_t st;  int k;
            if (kg < Kx) { sb = xt;    st = xStride; k = kg; }
            else         { sb = hprev; st = hStride; k = kg - Kx; }
#pragma unroll
            for (int rr = 0; rr < 2; ++rr) {
                const int row = rowA + rr * 32;
                async_ld16_(lds_off_(&Ach[h2][buf][row][colL]),
                            sb + (size_t)row * st + k);
            }
        }
    };

    // Accumulators for 4 M-tiles; half 0 carries the bias, half 1 starts at 0.
    v8f acc0, acc1, acc2, acc3;
    {
        const float bval = (hf == 0) ? bias[ncol] : 0.0f;
#pragma unroll
        for (int r = 0; r < 8; ++r) { acc0[r] = bval; acc1[r] = bval; acc2[r] = bval; acc3[r] = bval; }
    }

    issue(0, 0);
    for (int c = 0; c < nch; ++c) {
        const int cur = c & 1;
        if (c + 1 < nch) {
            issue(c + 1, cur ^ 1);                       // prefetch next chunk
            asm volatile("s_wait_asynccnt 0x4" ::: "memory");  // chunk c landed (in-order)
        } else {
            asm volatile("s_wait_asynccnt 0x0" ::: "memory");
        }
        __syncthreads();

        const int kbase = hf * Khalf + (c << 5);
        const float* Bb = (kbase < Kx) ? (W + (size_t)kbase * NG + ncol)
                                       : (R + (size_t)(kbase - Kx) * NG + ncol);
        const float (*A)[36] = Ach[hf][cur];
#pragma unroll
        for (int s = 0; s < 8; ++s) {
            const int kk = 4 * s + kksel;
            v2f bb;
            bb.x = Bb[(size_t)kk * NG];
            bb.y = Bb[(size_t)(kk + 1) * NG];
            v2f a0 = *(const v2f*)&A[ 0 + lc][kk];
            v2f a1 = *(const v2f*)&A[16 + lc][kk];
            v2f a2 = *(const v2f*)&A[32 + lc][kk];
            v2f a3 = *(const v2f*)&A[48 + lc][kk];
            acc0 = __builtin_amdgcn_wmma_f32_16x16x4_f32(false, a0, false, bb, (short)0, acc0, false, false);
            acc1 = __builtin_amdgcn_wmma_f32_16x16x4_f32(false, a1, false, bb, (short)0, acc1, false, false);
            acc2 = __builtin_amdgcn_wmma_f32_16x16x4_f32(false, a2, false, bb, (short)0, acc2, false, false);
            acc3 = __builtin_amdgcn_wmma_f32_16x16x4_f32(false, a3, false, bb, (short)0, acc3, false, false);
        }
        __syncthreads();   // also protects A buffer reuse and the smem union below
    }

    // Spill partial z to LDS column-major: zsh[half][g*16+lc][batch]
    {
        const int colWG = g * 16 + lc;
        const int mb = (lane >> 4) << 3;   // 0 or 8
        float* d0 = &zsh[hf][colWG][0];
#pragma unroll
        for (int r = 0; r < 8; ++r) d0[ 0 + mb + r] = acc0[r];
#pragma unroll
        for (int r = 0; r < 8; ++r) d0[16 + mb + r] = acc1[r];
#pragma unroll
        for (int r = 0; r < 8; ++r) d0[32 + mb + r] = acc2[r];
#pragma unroll
        for (int r = 0; r < 8; ++r) d0[48 + mb + r] = acc3[r];
    }
    __syncthreads();

    // Gate fusion: cols [0,16)=i, [16,32)=f, [32,48)=g, [48,64)=o ; sum the K-half partials
    for (int cell = tid; cell < 1024; cell += 256) {
        const int b  = cell & 63;
        const int uu = cell >> 6;          // 0..15
        const float zi = zsh[0][ 0 + uu][b] + zsh[1][ 0 + uu][b];
        const float zf = zsh[0][16 + uu][b] + zsh[1][16 + uu][b];
        const float zg = zsh[0][32 + uu][b] + zsh[1][32 + uu][b];
        const float zo = zsh[0][48 + uu][b] + zsh[1][48 + uu][b];
        const float ig = sigmoidf_(zi);
        const float fg = sigmoidf_(zf);
        const float og = sigmoidf_(zo);
        const int u = u0 + uu;
        const size_t ci = (size_t)b * UU + u;
        const float cp = first ? 0.0f : cstate[ci];
        const float cn = fg * cp + ig * zg;      // linear candidate (activation=None)
        cstate[ci] = cn;
        hout[(size_t)b * houtStride + u] = og * cn;  // linear output activation
    }
}

// Row-wise LayerNorm(eps=1e-3) + tanh over 1024-wide rows. One block per row.
__global__ __launch_bounds__(256)
void ln_tanh_kernel(const float* __restrict__ src, float* __restrict__ dst,
                    const float* __restrict__ gamma, const float* __restrict__ beta) {
    const int row = blockIdx.x;
    const float* s = src + (size_t)row * UU;
    float* d = dst + (size_t)row * UU;
    const int tid = threadIdx.x;

    float v[4];
    float sum = 0.0f, ssq = 0.0f;
#pragma unroll
    for (int j = 0; j < 4; ++j) {
        const float x = s[tid + j * 256];
        v[j] = x; sum += x; ssq += x * x;
    }
#pragma unroll
    for (int o = 16; o > 0; o >>= 1) {
        sum += __shfl_down(sum, o, 32);
        ssq += __shfl_down(ssq, o, 32);
    }
    __shared__ float rs[8], rq[8];
    const int wv = tid >> 5, lane = tid & 31;
    if (lane == 0) { rs[wv] = sum; rq[wv] = ssq; }
    __syncthreads();
    if (tid == 0) {
        float a = 0.0f, b = 0.0f;
#pragma unroll
        for (int i = 0; i < 8; ++i) { a += rs[i]; b += rq[i]; }
        rs[0] = a; rq[0] = b;
    }
    __syncthreads();
    const float mu   = rs[0] * (1.0f / 1024.0f);
    const float var  = rq[0] * (1.0f / 1024.0f) - mu * mu;
    const float rstd = rsqrtf(var + 1e-3f);
#pragma unroll
    for (int j = 0; j < 4; ++j) {
        const int c = tid + j * 256;
        d[c] = tanhf((v[j] - mu) * rstd * gamma[c] + beta[c]);
    }
}

extern "C" void kernel_launch(void* const* d_in, const int* in_sizes, int n_in,
                              void* d_out, int out_size, void* d_ws, size_t ws_size,
                              hipStream_t stream) {
    (void)in_sizes; (void)n_in; (void)out_size; (void)ws_size;
    const float* x      = (const float*)d_in[0];
    const float* W1     = (const float*)d_in[1];
    const float* R1     = (const float*)d_in[2];
    const float* b1     = (const float*)d_in[3];
    const float* gamma1 = (const float*)d_in[4];
    const float* beta1  = (const float*)d_in[5];
    const float* W2     = (const float*)d_in[6];
    const float* R2     = (const float*)d_in[7];
    const float* b2     = (const float*)d_in[8];
    const float* gamma2 = (const float*)d_in[9];
    const float* beta2  = (const float*)d_in[10];

    // Workspace: h1seq [B,T,U] (128 MB) | c1 [B,U] | c2 [B,U] | hbuf [2,B,U]
    float* h1 = (float*)d_ws;
    float* c1 = h1 + (size_t)BB * TT * UU;
    float* c2 = c1 + (size_t)BB * UU;
    float* hb = c2 + (size_t)BB * UU;

    const dim3 grd(64), blk(256);

    // Layer 1 recurrence: raw h_t written into h1seq (natural ping-pong over t)
    for (int t = 0; t < TT; ++t) {
        lstm_step_kernel<<<grd, blk, 0, stream>>>(
            x + (size_t)t * DD, (size_t)TT * DD, DD,
            t ? (h1 + (size_t)(t - 1) * UU) : nullptr, (size_t)TT * UU,
            W1, R1, b1, c1,
            h1 + (size_t)t * UU, (size_t)TT * UU, t == 0);
    }
    // LayerNorm + tanh in place over all B*T rows (layer-2 input)
    ln_tanh_kernel<<<BB * TT, 256, 0, stream>>>(h1, h1, gamma1, beta1);

    // Layer 2 recurrence: h ping-pong in hb
    for (int t = 0; t < TT; ++t) {
        const float* hp = hb + (size_t)(t & 1) * BB * UU;
        float*       ho = hb + (size_t)((t + 1) & 1) * BB * UU;
        lstm_step_kernel<<<grd, blk, 0, stream>>>(
            h1 + (size_t)t * UU, (size_t)TT * UU, UU,
            t ? hp : nullptr, (size_t)UU,
            W2, R2, b2, c2,
            ho, (size_t)UU, t == 0);
    }
    // Final h lives at parity (T & 1) == 0
    ln_tanh_kernel<<<BB, 256, 0, stream>>>(hb, (float*)d_out, gamma2, beta2);
}